// AdaptiveSimilarityLearning_16750372454915
// MI455X (gfx1250) — compile-verified
//
#include <hip/hip_runtime.h>

typedef float v2f __attribute__((ext_vector_type(2)));
typedef float v8f __attribute__((ext_vector_type(8)));

#define B_SZ 512
#define D_SZ 512
#define DH_SZ 256

// ---------------------------------------------------------------------------
// GEMM: C[M,N] = A[M,K] * W[N,K]^T   (both row-major, K contiguous)
// One wave computes a 16(M) x 64(N) strip using V_WMMA_F32_16X16X4_F32.
// fp32 WMMA fragment layout (wave32):
//   A 16x4 : lane l: row = l%16; VGPR v holds K = 2*(l/16) + v
//   B 4x16 : lane l: col = l%16; VGPR v holds K = 2*(l/16) + v
//   C 16x16: lane l: col = l%16; VGPR v holds row = v + 8*(l/16)
//
// K-loop is double-buffered at GROUP granularity (1 group = 4 k-steps = 16 K):
// two explicit register buffer sets (even/odd), hand-unrolled by 2 groups so
// all array indices are compile-time constants. While the 16 WMMAs of group g
// execute, the 20 loads of group g+1 are in flight.  Requires K % 32 == 0.
// ---------------------------------------------------------------------------

#define LOADG(g, A_, B_)                                                      \
    do {                                                                      \
        _Pragma("unroll") for (int s = 0; s < 4; ++s) {                       \
            const int kk = (g) * 16 + s * 4;                                  \
            A_[s]    = *(const v2f*)(aPtr  + kk);                             \
            B_[0][s] = *(const v2f*)(wPtr0 + kk);                             \
            B_[1][s] = *(const v2f*)(wPtr1 + kk);                             \
            B_[2][s] = *(const v2f*)(wPtr2 + kk);                             \
            B_[3][s] = *(const v2f*)(wPtr3 + kk);                             \
        }                                                                     \
    } while (0)

#define COMPUTEG(A_, B_)                                                      \
    do {                                                                      \
        _Pragma("unroll") for (int s = 0; s < 4; ++s) {                       \
            acc0 = __builtin_amdgcn_wmma_f32_16x16x4_f32(                     \
                false, A_[s], false, B_[0][s], (short)0, acc0, false, false); \
            acc1 = __builtin_amdgcn_wmma_f32_16x16x4_f32(                     \
                false, A_[s], false, B_[1][s], (short)0, acc1, false, false); \
            acc2 = __builtin_amdgcn_wmma_f32_16x16x4_f32(                     \
                false, A_[s], false, B_[2][s], (short)0, acc2, false, false); \
            acc3 = __builtin_amdgcn_wmma_f32_16x16x4_f32(                     \
                false, A_[s], false, B_[3][s], (short)0, acc3, false, false); \
        }                                                                     \
    } while (0)

__global__ __launch_bounds__(128) void gemm_ABt_wmma(
    const float* __restrict__ A, const float* __restrict__ W,
    float* __restrict__ C, int M, int N, int K)
{
    const int lane = threadIdx.x & 31;
    const int wave = threadIdx.x >> 5;
    const int gw   = blockIdx.x * 4 + wave;

    const int tilesN64 = N >> 6;                 // strips of 64 columns
    const int mTile = gw / tilesN64;
    const int nBase = (gw % tilesN64) << 6;
    if (mTile * 16 >= M) return;                 // wave-uniform (EXEC stays all-1)

    const int half = lane >> 4;                  // 0 or 1
    const int lm   = lane & 15;
    const int mRow = mTile * 16 + lm;

    const float* aPtr  = A + (size_t)mRow * K + 2 * half;
    const float* wPtr0 = W + (size_t)(nBase +  0 + lm) * K + 2 * half;
    const float* wPtr1 = W + (size_t)(nBase + 16 + lm) * K + 2 * half;
    const float* wPtr2 = W + (size_t)(nBase + 32 + lm) * K + 2 * half;
    const float* wPtr3 = W + (size_t)(nBase + 48 + lm) * K + 2 * half;

    v8f acc0 = {0.f,0.f,0.f,0.f,0.f,0.f,0.f,0.f};
    v8f acc1 = {0.f,0.f,0.f,0.f,0.f,0.f,0.f,0.f};
    v8f acc2 = {0.f,0.f,0.f,0.f,0.f,0.f,0.f,0.f};
    v8f acc3 = {0.f,0.f,0.f,0.f,0.f,0.f,0.f,0.f};

    // Two explicit buffer sets (even / odd group).
    v2f aE[4], bE[4][4];
    v2f aO[4], bO[4][4];

    const int G = K >> 4;                        // #groups of 16 K (G even)

    LOADG(0, aE, bE);                            // prologue
    for (int g = 0; g + 2 < G; g += 2) {
        LOADG(g + 1, aO, bO);                    // loads for odd group in flight
        COMPUTEG(aE, bE);                        // compute even group
        LOADG(g + 2, aE, bE);                    // loads for next even group
        COMPUTEG(aO, bO);                        // compute odd group
    }
    // epilogue: groups G-2 (in aE/bE) and G-1
    LOADG(G - 1, aO, bO);
    COMPUTEG(aE, bE);
    COMPUTEG(aO, bO);

    const int col0 = nBase + lm;
#pragma unroll
    for (int v = 0; v < 8; ++v) {
        const int m = mTile * 16 + v + 8 * half;
        float* crow = C + (size_t)m * N;
        crow[col0 +  0] = acc0[v];
        crow[col0 + 16] = acc1[v];
        crow[col0 + 32] = acc2[v];
        crow[col0 + 48] = acc3[v];
    }
}

// ---------------------------------------------------------------------------
// Pairwise ReLU-gated reduction + final combine (VALU-bound, L2-resident)
// ---------------------------------------------------------------------------
__device__ __forceinline__ float sigmoidf(float x) {
    return 1.0f / (1.0f + __expf(-x));
}

__global__ __launch_bounds__(256) void pairwise_kernel(
    const float* __restrict__ img_h, const float* __restrict__ txt_h,
    const float* __restrict__ img_t, const float* __restrict__ txt_t,
    const float* __restrict__ cosine,
    const float* __restrict__ b1,  const float* __restrict__ W2,  const float* __restrict__ b2p,
    const float* __restrict__ bt1, const float* __restrict__ Wt2, const float* __restrict__ bt2p,
    float* __restrict__ out)
{
    const int j = blockIdx.x * 16 + threadIdx.x;
    const int i = blockIdx.y * 16 + threadIdx.y;

    // similarity net: D = 512
    float acc_s = 0.f;
    {
        const float4* ai = (const float4*)(img_h + (size_t)i * D_SZ);
        const float4* bj = (const float4*)(txt_h + (size_t)j * D_SZ);
        const float4* bb = (const float4*)b1;
        const float4* ww = (const float4*)W2;
#pragma unroll 4
        for (int d = 0; d < D_SZ / 4; ++d) {
            float4 a = ai[d], b = bj[d], c = bb[d], w = ww[d];
            acc_s += fmaxf(a.x + b.x + c.x, 0.f) * w.x;
            acc_s += fmaxf(a.y + b.y + c.y, 0.f) * w.y;
            acc_s += fmaxf(a.z + b.z + c.z, 0.f) * w.z;
            acc_s += fmaxf(a.w + b.w + c.w, 0.f) * w.w;
        }
    }

    // temperature net: DH = 256
    float acc_t = 0.f;
    {
        const float4* ai = (const float4*)(img_t + (size_t)i * DH_SZ);
        const float4* bj = (const float4*)(txt_t + (size_t)j * DH_SZ);
        const float4* bb = (const float4*)bt1;
        const float4* ww = (const float4*)Wt2;
#pragma unroll 4
        for (int d = 0; d < DH_SZ / 4; ++d) {
            float4 a = ai[d], b = bj[d], c = bb[d], w = ww[d];
            acc_t += fmaxf(a.x + b.x + c.x, 0.f) * w.x;
            acc_t += fmaxf(a.y + b.y + c.y, 0.f) * w.y;
            acc_t += fmaxf(a.z + b.z + c.z, 0.f) * w.z;
            acc_t += fmaxf(a.w + b.w + c.w, 0.f) * w.w;
        }
    }

    const float adaptive_sim  = sigmoidf(acc_s + b2p[0]);
    const float adaptive_temp = 0.01f + 0.2f * sigmoidf(acc_t + bt2p[0]);
    const float cs = cosine[(size_t)i * B_SZ + j];

    out[(size_t)i * B_SZ + j] = (0.7f * cs + 0.3f * adaptive_sim) / adaptive_temp;
}

// ---------------------------------------------------------------------------
// Launch
// ---------------------------------------------------------------------------
extern "C" void kernel_launch(void* const* d_in, const int* in_sizes, int n_in,
                              void* d_out, int out_size, void* d_ws, size_t ws_size,
                              hipStream_t stream)
{
    const float* img_emb = (const float*)d_in[0];
    const float* txt_emb = (const float*)d_in[1];
    const float* W1a     = (const float*)d_in[2];
    const float* W1b     = (const float*)d_in[3];
    const float* b1      = (const float*)d_in[4];
    const float* W2      = (const float*)d_in[5];
    const float* b2      = (const float*)d_in[6];
    const float* Wt1a    = (const float*)d_in[7];
    const float* Wt1b    = (const float*)d_in[8];
    const float* bt1     = (const float*)d_in[9];
    const float* Wt2     = (const float*)d_in[10];
    const float* bt2     = (const float*)d_in[11];
    float* out = (float*)d_out;

    // Workspace layout (floats): img_h | txt_h | img_t | txt_t | cosine  = 4MB
    float* ws     = (float*)d_ws;
    float* img_h  = ws;
    float* txt_h  = img_h + (size_t)B_SZ * D_SZ;
    float* img_t  = txt_h + (size_t)B_SZ * D_SZ;
    float* txt_t  = img_t + (size_t)B_SZ * DH_SZ;
    float* cosine = txt_t + (size_t)B_SZ * DH_SZ;

    // GEMM launches: waves = (M/16)*(N/64), 4 waves (128 threads) per block.
    {
        int waves  = (B_SZ / 16) * (D_SZ / 64);          // 256
        gemm_ABt_wmma<<<waves / 4, 128, 0, stream>>>(img_emb, W1a, img_h, B_SZ, D_SZ, D_SZ);
        gemm_ABt_wmma<<<waves / 4, 128, 0, stream>>>(txt_emb, W1b, txt_h, B_SZ, D_SZ, D_SZ);
    }
    {
        int waves  = (B_SZ / 16) * (DH_SZ / 64);         // 128
        gemm_ABt_wmma<<<waves / 4, 128, 0, stream>>>(img_emb, Wt1a, img_t, B_SZ, DH_SZ, D_SZ);
        gemm_ABt_wmma<<<waves / 4, 128, 0, stream>>>(txt_emb, Wt1b, txt_t, B_SZ, DH_SZ, D_SZ);
    }
    {
        int waves  = (B_SZ / 16) * (B_SZ / 64);          // 256
        gemm_ABt_wmma<<<waves / 4, 128, 0, stream>>>(img_emb, txt_emb, cosine, B_SZ, B_SZ, D_SZ);
    }

    dim3 blk(16, 16);
    dim3 grd(B_SZ / 16, B_SZ / 16);
    pairwise_kernel<<<grd, blk, 0, stream>>>(img_h, txt_h, img_t, txt_t, cosine,
                                             b1, W2, b2, bt1, Wt2, bt2, out);
}